// NeuroDCG_39341900431843
// MI455X (gfx1250) — compile-verified
//
#include <hip/hip_runtime.h>
#include <hip/hip_bf16.h>

typedef __attribute__((ext_vector_type(2))) float vf2;
typedef __attribute__((ext_vector_type(8))) float vf8;

#define FIN   128
#define HC    128   // HEADS*OUT_C
#define NEG_SLOPE 0.2f

// ---------- helpers ----------
__device__ __forceinline__ float4 ld4(const float* p) {
    return *reinterpret_cast<const float4*>(p);
}
// order-preserving float -> uint encoding (for atomicMax-based segment max)
__device__ __forceinline__ unsigned fenc(float f) {
    unsigned u = __float_as_uint(f);
    return (u & 0x80000000u) ? ~u : (u | 0x80000000u);
}
__device__ __forceinline__ float fdec(unsigned u) {
    u = (u & 0x80000000u) ? (u & 0x7fffffffu) : ~u;
    return __uint_as_float(u);
}

// ---------- K1: dual GEMM  x@[W_l|W_r] + [b_l|b_r]  via V_WMMA_F32_16X16X4_F32 ----------
// block = 256 threads (8 waves). Each block: one 16-row M-tile, 16 column-tiles of 16
// (8 for W_l, 8 for W_r); wave w handles column tiles 2w and 2w+1.
__global__ void __launch_bounds__(256)
gat_gemm_wmma(const float* __restrict__ x,
              const float* __restrict__ Wl, const float* __restrict__ bl,
              const float* __restrict__ Wr, const float* __restrict__ br,
              float* __restrict__ xl, float* __restrict__ xr, int N)
{
    __shared__ float sA[16 * 132];           // padded: stride 132 -> conflict-free frags
    const int row0 = blockIdx.x * 16;
    const int tid  = threadIdx.x;

    // cooperative load of the 16x128 A tile (coalesced)
    for (int i = tid; i < 16 * 128; i += 256) {
        int r = i >> 7, c = i & 127;
        int gr = row0 + r;
        sA[r * 132 + c] = (gr < N) ? x[(size_t)gr * FIN + c] : 0.0f;
    }
    __syncthreads();

    const int wave = tid >> 5;
    const int lane = tid & 31;
    const int m    = lane & 15;        // A row / B,C,D column within tile
    const int half = lane >> 4;        // 0 | 1
    const int koff = half * 2;

    for (int t = 0; t < 2; ++t) {
        const int ct = wave * 2 + t;                     // 0..15
        const float* W  = (ct < 8) ? Wl : Wr;
        const float* bv = (ct < 8) ? bl : br;
        float* outp     = (ct < 8) ? xl : xr;
        const int col0  = (ct & 7) * 16;

        vf8 acc = {};
        #pragma unroll 8
        for (int k0 = 0; k0 < FIN; k0 += 4) {
            vf2 a, b;
            a.x = sA[m * 132 + k0 + koff];
            a.y = sA[m * 132 + k0 + koff + 1];
            b.x = W[(size_t)(k0 + koff)     * HC + col0 + m];
            b.y = W[(size_t)(k0 + koff + 1) * HC + col0 + m];
            acc = __builtin_amdgcn_wmma_f32_16x16x4_f32(
                      false, a, false, b, (short)0, acc, false, false);
        }
        const float bb = bv[col0 + m];
        #pragma unroll
        for (int v = 0; v < 8; ++v) {
            int mrow = row0 + v + 8 * half;
            if (mrow < N)
                outp[(size_t)mrow * HC + col0 + m] = acc[v] + bb;
        }
    }
}

// ---------- K2: per-edge GATv2 score + segment max ----------
// one wave per edge; lane l owns channels 4l..4l+3 (head = lane>>3)
__global__ void __launch_bounds__(256)
gat_edge_score(const int* __restrict__ src, const int* __restrict__ dst,
               const float* __restrict__ xl, const float* __restrict__ xr,
               const float* __restrict__ att,           // flat [128] == [h*32+c]
               float* __restrict__ alpha,               // [E*4] raw scores
               unsigned* __restrict__ amax_enc,         // [N*4] encoded max
               int E)
{
    const int e    = (blockIdx.x * blockDim.x + threadIdx.x) >> 5;
    const int lane = threadIdx.x & 31;
    if (e >= E) return;

    const int s = src[e], d = dst[e];
    const float4 a  = ld4(xl + (size_t)s * HC + 4 * lane);
    const float4 b  = ld4(xr + (size_t)d * HC + 4 * lane);
    const float4 av = ld4(att + 4 * lane);

    float tx = a.x + b.x, ty = a.y + b.y, tz = a.z + b.z, tw = a.w + b.w;
    tx = fmaxf(tx, NEG_SLOPE * tx);  ty = fmaxf(ty, NEG_SLOPE * ty);
    tz = fmaxf(tz, NEG_SLOPE * tz);  tw = fmaxf(tw, NEG_SLOPE * tw);

    float p = fmaf(tx, av.x, fmaf(ty, av.y, fmaf(tz, av.z, tw * av.w)));
    // butterfly reduce over the 8 lanes of each head
    p += __shfl_xor(p, 1);
    p += __shfl_xor(p, 2);
    p += __shfl_xor(p, 4);

    const int h = lane >> 3;
    if ((lane & 7) == 0) {
        alpha[(size_t)e * 4 + h] = p;
        atomicMax(&amax_enc[(size_t)d * 4 + h], fenc(p));
    }
}

// ---------- K3: exp(alpha - amax[dst]) + segment sum of denom ----------
__global__ void __launch_bounds__(256)
gat_edge_exp(const int* __restrict__ dst,
             float* __restrict__ alpha,                 // in: raw, out: expa
             const unsigned* __restrict__ amax_enc,
             float* __restrict__ denom, int E)
{
    const int i = blockIdx.x * blockDim.x + threadIdx.x;
    if (i >= E * 4) return;
    const int e = i >> 2, h = i & 3;
    const int d = dst[e];
    const float m  = fdec(amax_enc[(size_t)d * 4 + h]);
    const float ex = __expf(alpha[i] - m);
    alpha[i] = ex;
    unsafeAtomicAdd(&denom[(size_t)d * 4 + h], ex);
}

// ---------- K4: scatter-aggregate  out[dst] += expa * x_l[src] ----------
__global__ void __launch_bounds__(256)
gat_edge_aggregate(const int* __restrict__ src, const int* __restrict__ dst,
                   const float* __restrict__ xl, const float* __restrict__ expa,
                   float* __restrict__ out, int E)
{
    const int e    = (blockIdx.x * blockDim.x + threadIdx.x) >> 5;
    const int lane = threadIdx.x & 31;
    if (e >= E) return;

    const int s = src[e], d = dst[e];
    const float w  = expa[(size_t)e * 4 + (lane >> 3)];
    const float4 v = ld4(xl + (size_t)s * HC + 4 * lane);
    float* o = out + (size_t)d * HC + 4 * lane;
    unsafeAtomicAdd(o + 0, w * v.x);
    unsafeAtomicAdd(o + 1, w * v.y);
    unsafeAtomicAdd(o + 2, w * v.z);
    unsafeAtomicAdd(o + 3, w * v.w);
}

// ---------- K5: normalize by denom and add bias ----------
__global__ void __launch_bounds__(256)
gat_finalize(float* __restrict__ out, const float* __restrict__ denom,
             const float* __restrict__ bias, int N)
{
    const int i = blockIdx.x * blockDim.x + threadIdx.x;
    if (i >= N * HC) return;
    const int n = i >> 7, c = i & 127, h = c >> 5;
    out[i] = out[i] / (denom[(size_t)n * 4 + h] + 1e-16f) + bias[c];
}

// ---------- launcher ----------
extern "C" void kernel_launch(void* const* d_in, const int* in_sizes, int n_in,
                              void* d_out, int out_size, void* d_ws, size_t ws_size,
                              hipStream_t stream)
{
    (void)n_in; (void)out_size; (void)ws_size;
    const float* x    = (const float*)d_in[0];
    const int*   eidx = (const int*)  d_in[1];
    const float* Wl   = (const float*)d_in[2];
    const float* bl   = (const float*)d_in[3];
    const float* Wr   = (const float*)d_in[4];
    const float* br   = (const float*)d_in[5];
    const float* att  = (const float*)d_in[6];
    const float* bias = (const float*)d_in[7];
    float* out = (float*)d_out;

    const int N = in_sizes[0] / FIN;      // 50000
    const int E = in_sizes[1] / 2;        // 800000
    const int* src = eidx;
    const int* dst = eidx + E;

    // workspace layout
    float* xl    = (float*)d_ws;
    float* xr    = xl    + (size_t)N * HC;
    float* alpha = xr    + (size_t)N * HC;       // E*4 (raw scores, then expa)
    float* denom = alpha + (size_t)E * 4;        // N*4
    unsigned* amax_enc = (unsigned*)(denom + (size_t)N * 4);  // N*4

    // zero accumulators (deterministic per call; capture-safe)
    hipMemsetAsync(out,      0, (size_t)N * HC * sizeof(float), stream);
    hipMemsetAsync(denom,    0, (size_t)N * 4  * sizeof(float), stream);
    hipMemsetAsync(amax_enc, 0, (size_t)N * 4  * sizeof(unsigned), stream);

    // K1: dual GEMM via f32 WMMA
    gat_gemm_wmma<<<(N + 15) / 16, 256, 0, stream>>>(x, Wl, bl, Wr, br, xl, xr, N);

    // K2/K4: one wave per edge -> 8 edges per 256-thread block
    const int eblocks = (E + 7) / 8;
    gat_edge_score<<<eblocks, 256, 0, stream>>>(src, dst, xl, xr, att, alpha, amax_enc, E);

    gat_edge_exp<<<(E * 4 + 255) / 256, 256, 0, stream>>>(dst, alpha, amax_enc, denom, E);

    gat_edge_aggregate<<<eblocks, 256, 0, stream>>>(src, dst, xl, alpha, out, E);

    gat_finalize<<<((size_t)N * HC + 255) / 256, 256, 0, stream>>>(out, denom, bias, N);
}